// Net_82222853915381
// MI455X (gfx1250) — compile-verified
//
// CDNA5 / gfx1250 implementation of the gauge-equivariant mesh-CNN forward.
//
// Pipeline (all fp32, matching reference numerics):
//   1. build_g_kernel  : constructs G[q=10][Nout][Kin] on device (basis enum
//                        replicated from build_bases; no host copies).
//   2. conv_kernel<KD,ND,ORDER> : per-edge WMMA GEMM.
//        msg[e,o] = sum_q sum_i pc[e,q] * G[q,o,i] * xt[e,i]
//        - 4 waves/block, 1 wave = 16 edges (M-tile). Transported features
//          (order-2 rotation by edge connection) staged in LDS per wave.
//        - q loop: G[q] slice (<=80x80, 25.6KB) staged in LDS per workgroup,
//          barriers; K-loop in steps of 4 with v_wmma_f32_16x16x4_f32,
//          A pre-scaled by pc[e,q]. Accumulators live in VGPRs across all
//          K=10*Kin.
//        - epilogue: global_atomic_add_f32 scatter to agg[dst].
//   3. lin2_kernel     : 8192x8192x256 fp32 WMMA GEMM, 128x128 block tile,
//                        LDS staging of A/B K-chunks, +bias epilogue.
//   4. logsoftmax_kernel: per-row max / sum-exp / normalize (row = 32KB,
//                        L2-resident between the 3 passes).
// Elementwise glue (regular nonlinearity at 7 samples, residuals, lin1) are
// trivially bandwidth-bound VALU kernels.
//
// Workspace: ~17.6 MB of d_ws. Note: fp32 atomic scatter order is hardware-
// scheduled; bitwise-identical replays would need a dst-sorted segment sum.

#include <hip/hip_runtime.h>
#include <math.h>

typedef __attribute__((ext_vector_type(2))) float v2f;
typedef __attribute__((ext_vector_type(8))) float v8f;

#define N_NODES 8192
#define N_EDGES 57344

// Device-pass-only sanity check (host pass lacks the amdgcn builtin but
// still defines __AMDGCN__ under hipcc, so key on __HIP_DEVICE_COMPILE__).
#if defined(__HIP_DEVICE_COMPILE__)
#if !__has_builtin(__builtin_amdgcn_wmma_f32_16x16x4_f32)
#error "gfx1250: __builtin_amdgcn_wmma_f32_16x16x4_f32 not available"
#endif
#endif

// ---------------------------------------------------------------------------
// Equivariant basis enumeration (mirror of Python build_bases, band_limit=2).
// Returns the (r,k,O,od,I,id) entry of the expanded kernel G.
// ---------------------------------------------------------------------------
__device__ __forceinline__ float g_entry(const float* __restrict__ W, int R,
                                         int Co, int Ci, int n_in, int n_out,
                                         int r, int k, int O, int od, int I,
                                         int id) {
  const int BL = 2;
  float acc = 0.f;
  int b = 0;
  auto add = [&](int f, float a00, float a01, float a10, float a11, float c00,
                 float c01, float c10, float c11, int ar, int ac, int m,
                 int n) {
    int kc = (f == 0) ? 0 : 2 * f - 1;
    int ks = (f == 0) ? 0 : 2 * f;
    int ro = (m == 0) ? 0 : 2 * m - 1;
    int co = (n == 0) ? 0 : 2 * n - 1;
    int rr = od - ro, cc = id - co;
    float sel = 0.f;
    if (rr >= 0 && rr < ar && cc >= 0 && cc < ac) {
      float A2[2][2] = {{a00, a01}, {a10, a11}};
      float C2[2][2] = {{c00, c01}, {c10, c11}};
      if (k == kc) sel += A2[rr][cc];
      if (f > 0 && k == ks) sel += C2[rr][cc];
    }
    if (sel != 0.f) acc += W[((b * R + r) * Co + O) * Ci + I] * sel;
    ++b;
  };
  for (int n = 0; n <= n_in; ++n) {
    for (int m = 0; m <= n_out; ++m) {
      if (n == 0 && m == 0) {
        add(0, 1, 0, 0, 0, 0, 0, 0, 0, 1, 1, m, n);
      } else if (n == 0) {
        if (m <= BL) {
          add(m, 1, 0, 0, 0, 0, 0, 1, 0, 2, 1, m, n);
          add(m, 0, 0, 1, 0, -1, 0, 0, 0, 2, 1, m, n);
        }
      } else if (m == 0) {
        if (n <= BL) {
          add(n, 1, 0, 0, 0, 0, 1, 0, 0, 1, 2, m, n);
          add(n, 0, -1, 0, 0, 1, 0, 0, 0, 1, 2, m, n);
        }
      } else {
        int a = (m > n) ? (m - n) : (n - m);
        float s = (m > n) ? 1.f : ((m < n) ? -1.f : 0.f);
        if (a <= BL) {
          add(a, 1, 0, 0, 1, 0, -s, s, 0, 2, 2, m, n);
          add(a, 0, -1, 1, 0, -s, 0, 0, -s, 2, 2, m, n);
        }
        if (m + n <= BL) {
          add(m + n, 1, 0, 0, -1, 0, 1, 1, 0, 2, 2, m, n);
          add(m + n, 0, 1, 1, 0, -1, 0, 0, 1, 2, 2, m, n);
        }
      }
    }
  }
  return acc;
}

// G layout: [q=10][ND][KD], q = r*5+k, zero-padded outside (Co*dout, Ci*din).
__global__ __launch_bounds__(256) void build_g_kernel(
    const float* __restrict__ W, float* __restrict__ G, int Co, int Ci,
    int n_in, int n_out, int ND, int KD) {
  int idx = blockIdx.x * 256 + threadIdx.x;
  int total = 10 * ND * KD;
  if (idx >= total) return;
  int i_p = idx % KD;
  int o_p = (idx / KD) % ND;
  int rk = idx / (KD * ND);
  int din = 2 * n_in + 1, dout = 2 * n_out + 1;
  float v = 0.f;
  if (o_p < Co * dout && i_p < Ci * din) {
    v = g_entry(W, 2, Co, Ci, n_in, n_out, rk / 5, rk % 5, o_p / dout,
                o_p % dout, i_p / din, i_p % din);
  }
  G[idx] = v;
}

// ---------------------------------------------------------------------------
// Gauge-equivariant conv: per-edge WMMA GEMM + transport + pc-weighted q-sum
// + atomic scatter. KD = Ci*din (mult of 4), ND = Co*dout (mult of 16),
// ORDER = input rotation order (0 or 2; ORDER==2 implies din=5).
// ---------------------------------------------------------------------------
template <int KD, int ND, int ORDER>
__global__ __launch_bounds__(128) void conv_kernel(
    const float* __restrict__ x,     // [N_NODES, KD]
    const int* __restrict__ edges,   // [2, E] (row0 src, row1 dst)
    const float* __restrict__ pc,    // [E, 10]
    const float* __restrict__ conn,  // [E, 2]
    const float* __restrict__ G,     // [10, ND, KD]
    float* __restrict__ agg)         // [N_NODES, ND], pre-zeroed
{
  constexpr int KP = KD + 2;  // padded LDS row stride (bank-conflict break)
  constexpr int NT = ND / 16;
  __shared__ float Gs[ND * KP];
  __shared__ float xts[4][16 * KP];
  __shared__ float pcs[4][16 * 10];
  __shared__ int dsts[4][16];

  const int lane = threadIdx.x & 31;
  const int wave = threadIdx.x >> 5;
  const int e0 = (blockIdx.x * 4 + wave) * 16;  // this wave's 16 edges

  // --- stage transported features xt[16][KD] for this wave ---
  if (ORDER == 0) {
    for (int u = lane; u < 16 * KD; u += 32) {
      int m = u / KD, f = u % KD;
      int s = edges[e0 + m];
      xts[wave][m * KP + f] = x[s * KD + f];
    }
  } else {
    constexpr int CI = KD / 5;  // channels, din = 5
    for (int u = lane; u < 16 * CI; u += 32) {
      int m = u / CI, c = u % CI;
      int e = e0 + m;
      int s = edges[e];
      const float* xp = x + s * KD + c * 5;
      float f0 = xp[0], a1 = xp[1], bb = xp[2], a2 = xp[3], b2 = xp[4];
      float cg = conn[2 * e], sg = conn[2 * e + 1];
      float c2 = cg * cg - sg * sg, s2 = 2.f * cg * sg;
      float* o = &xts[wave][m * KP + c * 5];
      o[0] = f0;
      o[1] = cg * a1 - sg * bb;
      o[2] = sg * a1 + cg * bb;
      o[3] = c2 * a2 - s2 * b2;
      o[4] = s2 * a2 + c2 * b2;
    }
  }
  for (int u = lane; u < 160; u += 32) {  // pc[16 edges][10]
    int m = u / 10, q = u % 10;
    pcs[wave][m * 10 + q] = pc[(e0 + m) * 10 + q];
  }
  if (lane < 16) dsts[wave][lane] = edges[N_EDGES + e0 + lane];

  v8f acc[NT] = {};
  const int mrow = lane & 15;
  const int kb = (lane >> 4) << 1;  // K pair base per half-wave

  for (int q = 0; q < 10; ++q) {
    __syncthreads();
    for (int u = threadIdx.x; u < ND * KD; u += 128) {  // stage G[q]
      int o = u / KD, i = u % KD;
      Gs[o * KP + i] = G[(q * ND + o) * KD + i];
    }
    __syncthreads();
    const float pq = pcs[wave][mrow * 10 + q];
    for (int kk = 0; kk < KD; kk += 4) {
      v2f a;
      a.x = xts[wave][mrow * KP + kk + kb] * pq;
      a.y = xts[wave][mrow * KP + kk + kb + 1] * pq;
#pragma unroll
      for (int nt = 0; nt < NT; ++nt) {
        v2f bf;
        bf.x = Gs[(nt * 16 + mrow) * KP + kk + kb];
        bf.y = Gs[(nt * 16 + mrow) * KP + kk + kb + 1];
        acc[nt] = __builtin_amdgcn_wmma_f32_16x16x4_f32(
            false, a, false, bf, (short)0, acc[nt], false, false);
      }
    }
  }

  // --- scatter: D VGPR j holds (edge = j + 8*(lane/16), col = lane%16) ---
  const int mhalf = (lane >> 4) << 3;
#pragma unroll
  for (int nt = 0; nt < NT; ++nt) {
#pragma unroll
    for (int j = 0; j < 8; ++j) {
      int d = dsts[wave][j + mhalf];
      unsafeAtomicAdd(&agg[d * ND + nt * 16 + mrow], acc[nt][j]);
    }
  }
}

// ---------------------------------------------------------------------------
// Regular nonlinearity, order 2: sample at 7 angles, ReLU, project back.
// ---------------------------------------------------------------------------
__device__ __forceinline__ void regular_nl2(const float t[5], float o[5]) {
  float o0 = 0.f, o1 = 0.f, o2 = 0.f, o3 = 0.f, o4 = 0.f;
#pragma unroll
  for (int s = 0; s < 7; ++s) {
    float ang = 0.8975979010256552f * (float)s;  // 2*pi/7 * s
    float c1 = cosf(ang), s1 = sinf(ang);
    float c2 = c1 * c1 - s1 * s1, s2 = 2.f * c1 * s1;
    float f = t[0] + c1 * t[1] + s1 * t[2] + c2 * t[3] + s2 * t[4];
    float r = fmaxf(f, 0.f);
    o0 += r; o1 += c1 * r; o2 += s1 * r; o3 += c2 * r; o4 += s2 * r;
  }
  const float inv = 1.f / 7.f;
  o[0] = o0 * inv;
  o[1] = o1 * (2.f * inv);
  o[2] = o2 * (2.f * inv);
  o[3] = o3 * (2.f * inv);
  o[4] = o4 * (2.f * inv);
}

__global__ __launch_bounds__(256) void pad_pos_kernel(
    const float* __restrict__ pos, float* __restrict__ x0) {
  int idx = blockIdx.x * 256 + threadIdx.x;
  if (idx >= N_NODES * 16) return;
  int n = idx >> 4, c = idx & 15;
  x0[idx] = (c < 3) ? pos[n * 3 + c] : 0.f;
}

__global__ __launch_bounds__(256) void nl2_kernel(const float* __restrict__ in,
                                                  float* __restrict__ out) {
  int idx = blockIdx.x * 256 + threadIdx.x;
  if (idx >= N_NODES * 16) return;
  float t[5], o[5];
#pragma unroll
  for (int d = 0; d < 5; ++d) t[d] = in[idx * 5 + d];
  regular_nl2(t, o);
#pragma unroll
  for (int d = 0; d < 5; ++d) out[idx * 5 + d] = o[d];
}

// block-1 residual: h + pad(lin(pos)) then regular_nl2
__global__ __launch_bounds__(256) void b1_res_nl_kernel(
    const float* __restrict__ h2, const float* __restrict__ pos,
    const float* __restrict__ lin, float* __restrict__ out) {
  int idx = blockIdx.x * 256 + threadIdx.x;
  if (idx >= N_NODES * 16) return;
  int n = idx >> 4, c = idx & 15;
  float sc = lin[c * 3 + 0] * pos[n * 3 + 0] + lin[c * 3 + 1] * pos[n * 3 + 1] +
             lin[c * 3 + 2] * pos[n * 3 + 2];
  float t[5], o[5];
  t[0] = h2[idx * 5 + 0] + sc;  // shortcut only in d=0 (pad din=1 -> dout=5)
#pragma unroll
  for (int d = 1; d < 5; ++d) t[d] = h2[idx * 5 + d];
  regular_nl2(t, o);
#pragma unroll
  for (int d = 0; d < 5; ++d) out[idx * 5 + d] = o[d];
}

// block-2 residual: h + x then regular_nl2
__global__ __launch_bounds__(256) void add_nl2_kernel(
    const float* __restrict__ h2, const float* __restrict__ xres,
    float* __restrict__ out) {
  int idx = blockIdx.x * 256 + threadIdx.x;
  if (idx >= N_NODES * 16) return;
  float t[5], o[5];
#pragma unroll
  for (int d = 0; d < 5; ++d) t[d] = h2[idx * 5 + d] + xres[idx * 5 + d];
  regular_nl2(t, o);
#pragma unroll
  for (int d = 0; d < 5; ++d) out[idx * 5 + d] = o[d];
}

__global__ __launch_bounds__(256) void relu_kernel(const float* __restrict__ in,
                                                   float* __restrict__ out,
                                                   int total) {
  int idx = blockIdx.x * 256 + threadIdx.x;
  if (idx < total) out[idx] = fmaxf(in[idx], 0.f);
}

// block-3 residual: relu(h + x2[:, :, 0])   (dout=1 < din=5 truncation)
__global__ __launch_bounds__(256) void b3_res_relu_kernel(
    const float* __restrict__ h2, const float* __restrict__ x2,
    float* __restrict__ out) {
  int idx = blockIdx.x * 256 + threadIdx.x;
  if (idx >= N_NODES * 16) return;
  out[idx] = fmaxf(h2[idx] + x2[idx * 5], 0.f);
}

// lin1: [N,16] @ [256,16]^T + b, relu (tiny; VALU)
__global__ __launch_bounds__(256) void lin1_kernel(
    const float* __restrict__ x, const float* __restrict__ w,
    const float* __restrict__ b, float* __restrict__ out) {
  int idx = blockIdx.x * 256 + threadIdx.x;
  if (idx >= N_NODES * 256) return;
  int n = idx >> 8, o = idx & 255;
  float a = b[o];
#pragma unroll
  for (int k = 0; k < 16; ++k) a += x[n * 16 + k] * w[o * 16 + k];
  out[idx] = fmaxf(a, 0.f);
}

// ---------------------------------------------------------------------------
// lin2: out[8192,8192] = h[8192,256] @ w[8192,256]^T + bias   (fp32 WMMA)
// 128x128 block tile, 8 waves (each 16 rows x 128 cols), K staged in LDS.
// ---------------------------------------------------------------------------
__global__ __launch_bounds__(256) void lin2_kernel(
    const float* __restrict__ h, const float* __restrict__ w,
    const float* __restrict__ bias, float* __restrict__ out) {
  constexpr int KTOT = 256, KC = 32, KP = KC + 2;
  __shared__ float As[128 * KP];
  __shared__ float Bs[128 * KP];
  const int lane = threadIdx.x & 31;
  const int wave = threadIdx.x >> 5;
  const int m0 = blockIdx.x * 128;
  const int n0 = blockIdx.y * 128;
  v8f acc[8] = {};
  const int mrow = lane & 15;
  const int kb = (lane >> 4) << 1;
  for (int k0 = 0; k0 < KTOT; k0 += KC) {
    __syncthreads();
    for (int u = threadIdx.x; u < 128 * (KC / 4); u += 256) {
      int rrow = u / (KC / 4), c4 = (u % (KC / 4)) * 4;
      float4 va = *(const float4*)(h + (size_t)(m0 + rrow) * KTOT + k0 + c4);
      As[rrow * KP + c4 + 0] = va.x;
      As[rrow * KP + c4 + 1] = va.y;
      As[rrow * KP + c4 + 2] = va.z;
      As[rrow * KP + c4 + 3] = va.w;
      float4 vb = *(const float4*)(w + (size_t)(n0 + rrow) * KTOT + k0 + c4);
      Bs[rrow * KP + c4 + 0] = vb.x;
      Bs[rrow * KP + c4 + 1] = vb.y;
      Bs[rrow * KP + c4 + 2] = vb.z;
      Bs[rrow * KP + c4 + 3] = vb.w;
    }
    __syncthreads();
    for (int kk = 0; kk < KC; kk += 4) {
      v2f a;
      a.x = As[(wave * 16 + mrow) * KP + kk + kb];
      a.y = As[(wave * 16 + mrow) * KP + kk + kb + 1];
#pragma unroll
      for (int nt = 0; nt < 8; ++nt) {
        v2f bf;
        bf.x = Bs[(nt * 16 + mrow) * KP + kk + kb];
        bf.y = Bs[(nt * 16 + mrow) * KP + kk + kb + 1];
        acc[nt] = __builtin_amdgcn_wmma_f32_16x16x4_f32(
            false, a, false, bf, (short)0, acc[nt], false, false);
      }
    }
  }
  const int mhalf = (lane >> 4) << 3;
#pragma unroll
  for (int nt = 0; nt < 8; ++nt) {
#pragma unroll
    for (int j = 0; j < 8; ++j) {
      int m = m0 + wave * 16 + j + mhalf;
      int n = n0 + nt * 16 + mrow;
      out[(size_t)m * 8192 + n] = acc[nt][j] + bias[n];
    }
  }
}

// Row-wise log-softmax, in place: 3 passes over a 32KB row (L2-resident).
__global__ __launch_bounds__(256) void logsoftmax_kernel(
    float* __restrict__ out) {
  const int row = blockIdx.x;
  float* p = out + (size_t)row * 8192;
  __shared__ float red[256];
  float mx = -3.402823466e+38f;
  for (int j = threadIdx.x; j < 8192; j += 256) mx = fmaxf(mx, p[j]);
  red[threadIdx.x] = mx;
  __syncthreads();
  for (int s = 128; s > 0; s >>= 1) {
    if (threadIdx.x < s)
      red[threadIdx.x] = fmaxf(red[threadIdx.x], red[threadIdx.x + s]);
    __syncthreads();
  }
  mx = red[0];
  __syncthreads();
  float sum = 0.f;
  for (int j = threadIdx.x; j < 8192; j += 256) sum += expf(p[j] - mx);
  red[threadIdx.x] = sum;
  __syncthreads();
  for (int s = 128; s > 0; s >>= 1) {
    if (threadIdx.x < s) red[threadIdx.x] += red[threadIdx.x + s];
    __syncthreads();
  }
  float off = mx + logf(red[0]);
  for (int j = threadIdx.x; j < 8192; j += 256) p[j] -= off;
}

// ---------------------------------------------------------------------------
extern "C" void kernel_launch(void* const* d_in, const int* in_sizes, int n_in_,
                              void* d_out, int out_size, void* d_ws,
                              size_t ws_size, hipStream_t stream) {
  const float* pos = (const float*)d_in[0];
  const int* edges = (const int*)d_in[1];
  const float* pc = (const float*)d_in[2];
  const float* conn = (const float*)d_in[3];
  const float* w_b1c1 = (const float*)d_in[4];
  const float* w_b1c2 = (const float*)d_in[5];
  const float* b1lin = (const float*)d_in[6];
  const float* w_b2c1 = (const float*)d_in[7];
  const float* w_b2c2 = (const float*)d_in[8];
  const float* w_b3c1 = (const float*)d_in[9];
  const float* w_b3c2 = (const float*)d_in[10];
  const float* w1 = (const float*)d_in[11];
  const float* bb1 = (const float*)d_in[12];
  const float* w2 = (const float*)d_in[13];
  const float* bb2 = (const float*)d_in[14];
  float* out = (float*)d_out;

  // workspace carve-up (floats); total ~4.39M floats = 17.6 MB
  float* xA = (float*)d_ws;            // [N,80]
  float* xB = xA + N_NODES * 80;       // [N,80]
  float* xC = xB + N_NODES * 80;       // [N,80]
  float* G = xC + N_NODES * 80;        // [10,80,80] max
  float* t16a = G + 64000;             // [N,16]
  float* t16b = t16a + N_NODES * 16;   // [N,16]
  float* h256 = t16b + N_NODES * 16;   // [N,256]

  const dim3 cgrid(N_EDGES / 64), cblk(128);  // 16 edges/wave, 4 waves/block
  auto gb = [](int total) { return (total + 255) / 256; };
  const size_t sz80 = (size_t)N_NODES * 80 * sizeof(float);
  const size_t sz16 = (size_t)N_NODES * 16 * sizeof(float);

  // ---- block 1 (n_in=0 -> n_out=2) ----
  pad_pos_kernel<<<gb(N_NODES * 16), 256, 0, stream>>>(pos, t16a);
  build_g_kernel<<<gb(10 * 80 * 16), 256, 0, stream>>>(w_b1c1, G, 16, 3, 0, 2,
                                                       80, 16);
  (void)hipMemsetAsync(xA, 0, sz80, stream);
  conv_kernel<16, 80, 0><<<cgrid, cblk, 0, stream>>>(t16a, edges, pc, conn, G,
                                                     xA);
  nl2_kernel<<<gb(N_NODES * 16), 256, 0, stream>>>(xA, xB);
  build_g_kernel<<<gb(10 * 80 * 80), 256, 0, stream>>>(w_b1c2, G, 16, 16, 2, 2,
                                                       80, 80);
  (void)hipMemsetAsync(xA, 0, sz80, stream);
  conv_kernel<80, 80, 2><<<cgrid, cblk, 0, stream>>>(xB, edges, pc, conn, G,
                                                     xA);
  b1_res_nl_kernel<<<gb(N_NODES * 16), 256, 0, stream>>>(xA, pos, b1lin, xB);

  // ---- block 2 (2 -> 2), x1 in xB ----
  build_g_kernel<<<gb(10 * 80 * 80), 256, 0, stream>>>(w_b2c1, G, 16, 16, 2, 2,
                                                       80, 80);
  (void)hipMemsetAsync(xA, 0, sz80, stream);
  conv_kernel<80, 80, 2><<<cgrid, cblk, 0, stream>>>(xB, edges, pc, conn, G,
                                                     xA);
  nl2_kernel<<<gb(N_NODES * 16), 256, 0, stream>>>(xA, xC);
  build_g_kernel<<<gb(10 * 80 * 80), 256, 0, stream>>>(w_b2c2, G, 16, 16, 2, 2,
                                                       80, 80);
  (void)hipMemsetAsync(xA, 0, sz80, stream);
  conv_kernel<80, 80, 2><<<cgrid, cblk, 0, stream>>>(xC, edges, pc, conn, G,
                                                     xA);
  add_nl2_kernel<<<gb(N_NODES * 16), 256, 0, stream>>>(xA, xB, xC);

  // ---- block 3 (2 -> 0), x2 in xC ----
  build_g_kernel<<<gb(10 * 16 * 80), 256, 0, stream>>>(w_b3c1, G, 16, 16, 2, 0,
                                                       16, 80);
  (void)hipMemsetAsync(t16a, 0, sz16, stream);
  conv_kernel<80, 16, 2><<<cgrid, cblk, 0, stream>>>(xC, edges, pc, conn, G,
                                                     t16a);
  relu_kernel<<<gb(N_NODES * 16), 256, 0, stream>>>(t16a, t16b, N_NODES * 16);
  build_g_kernel<<<gb(10 * 16 * 16), 256, 0, stream>>>(w_b3c2, G, 16, 16, 0, 0,
                                                       16, 16);
  (void)hipMemsetAsync(t16a, 0, sz16, stream);
  conv_kernel<16, 16, 0><<<cgrid, cblk, 0, stream>>>(t16b, edges, pc, conn, G,
                                                     t16a);
  b3_res_relu_kernel<<<gb(N_NODES * 16), 256, 0, stream>>>(t16a, xC, t16b);

  // ---- head ----
  lin1_kernel<<<gb(N_NODES * 256), 256, 0, stream>>>(t16b, w1, bb1, h256);
  lin2_kernel<<<dim3(64, 64), 256, 0, stream>>>(h256, w2, bb2, out);
  logsoftmax_kernel<<<N_NODES, 256, 0, stream>>>(out);
}